// disp_loss_46849503265184
// MI455X (gfx1250) — compile-verified
//
#include <hip/hip_runtime.h>
#include <hip/hip_bf16.h>

#define LAYERS   4
#define BROWS    256
#define HDIM     65536
#define KC       64              // K chunk staged in LDS
#define SPLITK   64              // K-slices per (layer, half)
#define HALVES   2               // row halves (128 rows each)
#define LDST     68              // LDS row stride in floats (16B-aligned, bank-staggered)

typedef float v2f __attribute__((ext_vector_type(2)));
typedef float v8f __attribute__((ext_vector_type(8)));

// ---------------------------------------------------------------------------
// Zero the Gram workspace and the output scalar (harness poisons once and
// graph-replays without re-init, so this must run every call).
// ---------------------------------------------------------------------------
__global__ void dl_zero_kernel(float* __restrict__ G, float* __restrict__ out, int n) {
    int i = blockIdx.x * blockDim.x + threadIdx.x;
    if (i < n) G[i] = 0.0f;
    if (i == 0) out[0] = 0.0f;
}

// ---------------------------------------------------------------------------
// Split-K Gram: G[l] += H[l] * H[l]^T over this workgroup's K slice.
// One LDS buffer holds ALL 256 rows of the K chunk (A and B operands are the
// same data for a Gram matrix). 16 wave32 waves; each wave owns a 16x128
// output strip = 8 WMMA tiles, reusing its A fragment across all 8.
// ---------------------------------------------------------------------------
__global__ void __launch_bounds__(512)
dl_gram_kernel(const float* __restrict__ X, float* __restrict__ G) {
    int wg = blockIdx.x;
    const int half = wg % HALVES; wg /= HALVES;   // which 128-row half we produce
    const int s    = wg % SPLITK; wg /= SPLITK;   // K slice
    const int l    = wg;                          // layer

    __shared__ float lds[BROWS * LDST];           // 69,632 B

    const int tid  = threadIdx.x;
    const int lane = tid & 31;
    const int wv   = tid >> 5;        // 0..15
    const int ist  = wv & 7;          // row strip within half (16 rows each)
    const int ch   = wv >> 3;         // column half (128 cols each)

    const float* __restrict__ Xl = X + (size_t)l * BROWS * HDIM;

    v8f acc[8];
#pragma unroll
    for (int j = 0; j < 8; ++j) acc[j] = (v8f){0.f,0.f,0.f,0.f,0.f,0.f,0.f,0.f};

    // WMMA f32 16x16x4 per-lane layout: M/N = lane%16, K = (lane/16)*2 + {0,1}
    const int arow = half * 128 + ist * 16 + (lane & 15);
    const int brow = ch * 128 + (lane & 15);
    const int kpar = (lane >> 4) * 2;

    const int k0      = s * (HDIM / SPLITK);
    const int nchunks = (HDIM / SPLITK) / KC;     // 16

    for (int c = 0; c < nchunks; ++c) {
        const int kb = k0 + c * KC;
        __syncthreads();
        // Stage all 256 rows x KC floats: 4096 float4 / 512 threads = 8 each.
#pragma unroll
        for (int it = 0; it < 8; ++it) {
            const int f   = tid + it * 512;       // 0..4095
            const int row = f >> 4;
            const int kq  = f & 15;
            const float4 v4 = *(const float4*)(Xl + (size_t)row * HDIM + kb + kq * 4);
            *(float4*)(&lds[row * LDST + kq * 4]) = v4;
        }
        __syncthreads();

        const float* pa = &lds[arow * LDST + kpar];
        const float* pb = &lds[brow * LDST + kpar];
#pragma unroll 4
        for (int kk = 0; kk < KC; kk += 4) {
            const v2f a = *(const v2f*)(pa + kk);   // loaded once, reused 8x
#pragma unroll
            for (int j = 0; j < 8; ++j) {
                const v2f b = *(const v2f*)(pb + j * (16 * LDST) + kk);
                acc[j] = __builtin_amdgcn_wmma_f32_16x16x4_f32(
                    /*neg_a=*/false, a, /*neg_b=*/false, b,
                    /*c_mod=*/(short)0, acc[j],
                    /*reuse_a=*/false, /*reuse_b=*/false);
            }
        }
    }

    // C/D layout: VGPR v holds M = v + (lane/16)*8, N = lane%16.
    float* __restrict__ Gl = G + (size_t)l * BROWS * BROWS;
    const int col  = lane & 15;
    const int rowb = half * 128 + ist * 16 + (lane >> 4) * 8;
    const int colb = ch * 128 + col;
#pragma unroll
    for (int j = 0; j < 8; ++j) {
#pragma unroll
        for (int v = 0; v < 8; ++v) {
            atomicAdd(&Gl[(size_t)(rowb + v) * BROWS + colb + j * 16], acc[j][v]);
        }
    }
}

// ---------------------------------------------------------------------------
// Epilogue: from raw Gram G[l] recover normalized distances, compute the
// dispersion loss, reduce, atomically accumulate into out[0].
//   x = h / max(||h||, 1e-12)  =>  dot = G[b,c]/(n_b n_c),  sq_b = G[b,b]/n_b^2
// ---------------------------------------------------------------------------
__global__ void __launch_bounds__(256)
dl_loss_kernel(const float* __restrict__ G, const int* __restrict__ task,
               float* __restrict__ out) {
    const int l = blockIdx.x;
    const int b = threadIdx.x;

    __shared__ float inv_n[BROWS];
    __shared__ float sqv[BROWS];
    __shared__ int   tt[BROWS];
    __shared__ float red[BROWS];

    const float* __restrict__ Gl = G + (size_t)l * BROWS * BROWS;

    const float n2  = Gl[(size_t)b * BROWS + b];
    const float n   = fmaxf(sqrtf(fmaxf(n2, 0.0f)), 1e-12f);
    const float inv = 1.0f / n;
    inv_n[b] = inv;
    sqv[b]   = n2 * inv * inv;    // ||x_b||^2 after normalization
    tt[b]    = task[b];
    __syncthreads();

    const float sb = sqv[b];
    const int   tb = tt[b];
    float asum = 0.0f, bsum = 0.0f;
    for (int c = 0; c < BROWS; ++c) {
        const float dot = Gl[(size_t)b * BROWS + c] * inv * inv_n[c];
        const float d2  = fmaxf(sb + sqv[c] - 2.0f * dot, 0.0f);
        const float w   = __expf(-2.0f * d2);      // TEMPERATURE = 0.5
        bsum += w;
        if (tt[c] == tb && c != b) asum += w;
    }
    red[b] = -logf((asum + 1e-8f) / (bsum + 1e-8f));
    __syncthreads();

    for (int off = 128; off > 0; off >>= 1) {
        if (b < off) red[b] += red[b + off];
        __syncthreads();
    }
    if (b == 0) {
        // mean over B, sum over L / L, * LOSS_WEIGHT
        atomicAdd(out, red[0] * (0.2f / (256.0f * (float)LAYERS)));
    }
}

// ---------------------------------------------------------------------------
extern "C" void kernel_launch(void* const* d_in, const int* in_sizes, int n_in,
                              void* d_out, int out_size, void* d_ws, size_t ws_size,
                              hipStream_t stream) {
    const float* X    = (const float*)d_in[0];   // [L, B, H] f32
    const int*   task = (const int*)d_in[1];     // [B] int
    float*       out  = (float*)d_out;           // scalar f32
    float*       G    = (float*)d_ws;            // [L, B, B] f32 workspace (1 MB)

    const int gsize = LAYERS * BROWS * BROWS;
    dl_zero_kernel<<<(gsize + 255) / 256, 256, 0, stream>>>(G, out, gsize);

    dim3 grid(LAYERS * SPLITK * HALVES);         // 512 workgroups
    dl_gram_kernel<<<grid, 512, 0, stream>>>(X, G);

    dl_loss_kernel<<<LAYERS, BROWS, 0, stream>>>(G, task, out);
}